// GNNEncoder_89412629168562
// MI455X (gfx1250) — compile-verified
//
#include <hip/hip_runtime.h>

typedef __attribute__((ext_vector_type(2))) float v2f;
typedef __attribute__((ext_vector_type(8))) float v8f;

#define NNODES 50000
#define FEAT   96
#define MTILES 3125   /* 50000 / 16 */
#define NTILES 6      /* 96 / 16    */
#define KSTEPS 24     /* 96 / 4     */

// ---------------------------------------------------------------------------
// Zero out agg + deg scratch (N*96 + N floats)
// ---------------------------------------------------------------------------
__global__ void zero_f32(float* __restrict__ p, long long n) {
    long long i = (long long)blockIdx.x * blockDim.x + threadIdx.x;
    if (i < n) p[i] = 0.0f;
}

// ---------------------------------------------------------------------------
// Scatter-sum: one thread per (edge, 16-byte feature chunk).
// 24 chunks cover the 96-float feature row. Degree bumped by chunk 0.
// ---------------------------------------------------------------------------
__global__ void sage_scatter(const long long* __restrict__ ei,
                             const float* __restrict__ xin,
                             float* __restrict__ agg,
                             float* __restrict__ deg,
                             long long E) {
    long long tid = (long long)blockIdx.x * blockDim.x + threadIdx.x;
    long long total = E * 24;
    if (tid >= total) return;
    long long e = tid / 24;
    int c = (int)(tid - e * 24);

    // speculative prefetch of edge indices ahead (global_prefetch_b8)
    __builtin_prefetch(ei + e + 2048, 0, 1);

    int s = (int)ei[e];
    int d = (int)ei[E + e];

    const float4 v = *(const float4*)(xin + (long long)s * FEAT + c * 4);
    float* a = agg + (long long)d * FEAT + c * 4;
    atomicAdd(a + 0, v.x);
    atomicAdd(a + 1, v.y);
    atomicAdd(a + 2, v.z);
    atomicAdd(a + 3, v.w);
    if (c == 0) atomicAdd(deg + d, 1.0f);
}

// ---------------------------------------------------------------------------
// Fused: out = relu( (agg/deg) @ Wl + x @ Wr + b )
// One wave per 16-row strip; 6 WMMA accumulators cover all 96 columns.
// fp32 WMMA 16x16x4. Weights staged in LDS *pre-swizzled into fragment
// order* so every lane's B fragment is one contiguous 8-byte pair:
//   sW[((kk*6 + n)*32 + lane)*2 + {0,1}] = W[4kk + 2h + {0,1}][16n + m],
//   lane = h*16 + m  ->  inner loop is a single ds_load_b64 per fragment.
// ---------------------------------------------------------------------------
__global__ __launch_bounds__(256)
void sage_gemm(const float* __restrict__ agg,
               const float* __restrict__ deg,
               const float* __restrict__ xin,
               const float* __restrict__ Wl,
               const float* __restrict__ Wr,
               const float* __restrict__ bias,
               float* __restrict__ out) {
    __shared__ float sWl[FEAT * FEAT];
    __shared__ float sWr[FEAT * FEAT];
    __shared__ float sb[FEAT];

    // Cooperative swizzled stage of both weight matrices into LDS.
    {
        const float4* gl = (const float4*)Wl;
        const float4* gr = (const float4*)Wr;
        for (int i = threadIdx.x; i < (FEAT * FEAT) / 4; i += 256) {
            const float4 vl = gl[i];
            const float4 vr = gr[i];
            const int base = i * 4;
            const int k  = base / FEAT;        // 4 elems share one k (96 % 4 == 0)
            const int c0 = base - k * FEAT;
            const int kk = k >> 2;
            const int j  = k & 3;
            const int hl = (j >> 1) << 4;      // lane-half bit
            const int jb = j & 1;              // which element of the pair
            const float el[4] = {vl.x, vl.y, vl.z, vl.w};
            const float er[4] = {vr.x, vr.y, vr.z, vr.w};
#pragma unroll
            for (int q = 0; q < 4; ++q) {
                const int cc = c0 + q;
                const int n = cc >> 4;
                const int m = cc & 15;
                const int dst = (((kk * NTILES + n) << 5) + (hl | m)) * 2 + jb;
                sWl[dst] = el[q];
                sWr[dst] = er[q];
            }
        }
        if (threadIdx.x < FEAT) sb[threadIdx.x] = bias[threadIdx.x];
    }
    __syncthreads();

    const int wave = threadIdx.x >> 5;
    const int lane = threadIdx.x & 31;
    const int strip = blockIdx.x * 8 + wave;
    if (strip >= MTILES) return;   // wave-uniform exit; EXEC all-ones inside

    const int m = lane & 15;       // row within tile for A; col within tile for B/C
    const int h = lane >> 4;       // lane-half selects K offset (A/B) / row+8 (C)
    const int mbase = strip * 16;

    // fused mean-normalization: per-row 1/max(deg,1)
    const float dv = deg[mbase + m];
    const float scale = 1.0f / fmaxf(dv, 1.0f);

    // init accumulators with bias (same for every row of the tile)
    v8f c[NTILES];
#pragma unroll
    for (int n = 0; n < NTILES; ++n) {
        const float b0 = sb[16 * n + m];
#pragma unroll
        for (int r = 0; r < 8; ++r) c[n][r] = b0;
    }

    const float* arow = agg + (long long)(mbase + m) * FEAT;
    const float* xrow = xin + (long long)(mbase + m) * FEAT;
    const v2f* fWl = (const v2f*)sWl;
    const v2f* fWr = (const v2f*)sWr;

#pragma unroll 4
    for (int kk = 0; kk < KSTEPS; ++kk) {
        // A fragments: contiguous 8-byte pair per lane (global_load_b64)
        const v2f ta = *(const v2f*)(arow + 4 * kk + 2 * h);
        const v2f tx = *(const v2f*)(xrow + 4 * kk + 2 * h);
        v2f aA;
        aA.x = scale * ta.x;
        aA.y = scale * ta.y;

        const int fbase = (kk * NTILES) * 32 + lane;
#pragma unroll
        for (int n = 0; n < NTILES; ++n) {
            // B fragments: one ds_load_b64 each, already in WMMA layout
            const v2f bL = fWl[fbase + n * 32];
            const v2f bR = fWr[fbase + n * 32];

            c[n] = __builtin_amdgcn_wmma_f32_16x16x4_f32(
                false, aA, false, bL, (short)0, c[n], false, false);
            c[n] = __builtin_amdgcn_wmma_f32_16x16x4_f32(
                false, tx, false, bR, (short)0, c[n], false, false);
        }
    }

    // Store with ReLU. C layout: VGPR r -> row r (lanes 0-15) / row r+8 (16-31)
#pragma unroll
    for (int n = 0; n < NTILES; ++n) {
        const int col = 16 * n + m;
#pragma unroll
        for (int r = 0; r < 8; ++r) {
            const int row = mbase + r + 8 * h;
            out[(long long)row * FEAT + col] = fmaxf(c[n][r], 0.0f);
        }
    }
}

// ---------------------------------------------------------------------------
// Launch: two SAGE layers.
// ws layout: agg[N*96] | deg[N] | h1[N*96]
// ---------------------------------------------------------------------------
extern "C" void kernel_launch(void* const* d_in, const int* in_sizes, int n_in,
                              void* d_out, int out_size, void* d_ws, size_t ws_size,
                              hipStream_t stream) {
    const float*     x   = (const float*)d_in[0];
    const long long* ei  = (const long long*)d_in[1];
    const float*     Wl0 = (const float*)d_in[2];
    const float*     Wr0 = (const float*)d_in[3];
    const float*     b0  = (const float*)d_in[4];
    const float*     Wl1 = (const float*)d_in[5];
    const float*     Wr1 = (const float*)d_in[6];
    const float*     b1  = (const float*)d_in[7];
    float*           out = (float*)d_out;

    const long long E = (long long)in_sizes[1] / 2;
    const long long N = NNODES;

    float* agg = (float*)d_ws;
    float* deg = agg + (size_t)N * FEAT;
    float* h1  = deg + (size_t)N;

    const long long zn = N * (FEAT + 1);           // agg + deg contiguous
    const int zblocks = (int)((zn + 255) / 256);
    const int sblocks = (int)((E * 24 + 255) / 256);
    const int gblocks = (MTILES + 7) / 8;

    // ---- layer 1 ----
    zero_f32<<<zblocks, 256, 0, stream>>>(agg, zn);
    sage_scatter<<<sblocks, 256, 0, stream>>>(ei, x, agg, deg, E);
    sage_gemm<<<gblocks, 256, 0, stream>>>(agg, deg, x, Wl0, Wr0, b0, h1);

    // ---- layer 2 ----
    zero_f32<<<zblocks, 256, 0, stream>>>(agg, zn);
    sage_scatter<<<sblocks, 256, 0, stream>>>(ei, h1, agg, deg, E);
    sage_gemm<<<gblocks, 256, 0, stream>>>(agg, deg, h1, Wl1, Wr1, b1, out);
}